// TransformerBlock_85830626443902
// MI455X (gfx1250) — compile-verified
//
#include <hip/hip_runtime.h>

#define BB    64
#define NN    1024
#define KNB   20        // K neighbors
#define KP    32        // padded neighbor count
#define HEADS 8
#define DIMK  32
#define DIMV  32
#define CH    64        // CH_RAISE
#define CC    40        // C_CAT
#define TP    2         // points per workgroup
#define NCOL  (TP*KP)   // 64 columns per workgroup

typedef float v2f __attribute__((ext_vector_type(2)));
typedef float v8f __attribute__((ext_vector_type(8)));

__device__ __forceinline__ v8f wmma4(v2f a, v2f b, v8f c) {
    // D(16x16,f32) += A(16x4,f32) * B(4x16,f32)
    return __builtin_amdgcn_wmma_f32_16x16x4_f32(
        /*neg_a=*/false, a, /*neg_b=*/false, b,
        /*c_mod=*/(short)0, c, /*reuse_a=*/false, /*reuse_b=*/false);
}

// ---------------------------------------------------------------------------
// Kernel 1: passthrough xyz -> first output slot
// ---------------------------------------------------------------------------
__global__ __launch_bounds__(256) void copy_xyz_kernel(const float4* __restrict__ in,
                                                       float4* __restrict__ out, int n4) {
    int i = blockIdx.x * 256 + threadIdx.x;
    if (i < n4) out[i] = in[i];
}

// ---------------------------------------------------------------------------
// Kernel 2: per-batch 4-D KNN (top-20 smallest squared distance, stable ties)
// ---------------------------------------------------------------------------
__global__ __launch_bounds__(256) void knn_kernel(const float* __restrict__ x,
                                                  int* __restrict__ knn) {
    __shared__ float4 sxyz[NN];                     // 16 KB
    const int b    = blockIdx.x / (NN / 256);
    const int tile = blockIdx.x % (NN / 256);
    const float4* xb = (const float4*)(x + (size_t)b * NN * 4);
    for (int i = threadIdx.x; i < NN; i += 256) sxyz[i] = xb[i];
    __syncthreads();

    const int n = tile * 256 + threadIdx.x;
    const float4 c = sxyz[n];
    float bd[KNB]; int bi[KNB];
#pragma unroll
    for (int j = 0; j < KNB; ++j) { bd[j] = 3.4e38f; bi[j] = 0; }

    for (int m = 0; m < NN; ++m) {
        float4 q = sxyz[m];
        float dx = q.x - c.x, dy = q.y - c.y, dz = q.z - c.z, dw = q.w - c.w;
        float d = dx * dx + dy * dy + dz * dz + dw * dw;
        if (d < bd[KNB - 1]) {
            bd[KNB - 1] = d; bi[KNB - 1] = m;
#pragma unroll
            for (int j = KNB - 1; j > 0; --j) {
                if (bd[j] < bd[j - 1]) {
                    float td = bd[j]; bd[j] = bd[j - 1]; bd[j - 1] = td;
                    int   ti = bi[j]; bi[j] = bi[j - 1]; bi[j - 1] = ti;
                }
            }
        }
    }
    int* op = knn + ((size_t)b * NN + n) * KNB;
#pragma unroll
    for (int j = 0; j < KNB; ++j) op[j] = bi[j];
}

// ---------------------------------------------------------------------------
// Kernel 3: fused feature/MLP/attention pipeline, one wave32 per 2 points
// ---------------------------------------------------------------------------
__global__ __launch_bounds__(32) void fused_kernel(
    const float* __restrict__ x,  const int* __restrict__ knn,
    const float* __restrict__ Wh, const float* __restrict__ W1,
    const float* __restrict__ g1, const float* __restrict__ b1,
    const float* __restrict__ W2, const float* __restrict__ g2, const float* __restrict__ b2,
    const float* __restrict__ Wv, const float* __restrict__ gv, const float* __restrict__ bv,
    const float* __restrict__ Wk, const float* __restrict__ Wq,
    const float* __restrict__ gq, const float* __restrict__ bq,
    float* __restrict__ out)
{
    __shared__ float sFeat[CC][NCOL];        // 10 KB
    __shared__ float sH   [CH][NCOL];        // 16 KB (h1)
    __shared__ float sV   [DIMV][NCOL];      //  8 KB
    __shared__ float sK   [DIMK][NCOL];      //  8 KB
    __shared__ float sKV  [TP][DIMK * DIMV]; //  8 KB
    __shared__ float sFQ  [CH][16];          //  4 KB (cols >= TP stay zero)
    __shared__ float sQ   [HEADS * DIMK][TP];//  2 KB

    const int lane  = threadIdx.x;
    const int b     = blockIdx.x / (NN / TP);
    const int nbase = (blockIdx.x % (NN / TP)) * TP;
    const int mrow  = lane & 15;
    const int khalf = (lane >> 4) * 2;       // K offset within 4-step for this half

    __builtin_prefetch(W1, 0, 3);
    __builtin_prefetch(W2, 0, 3);
    __builtin_prefetch(Wq, 0, 3);

    for (int i = lane; i < CH * 16; i += 32) (&sFQ[0][0])[i] = 0.0f;

    // ---- stage 1: build feature (40 x NCOL): [gnorm(4); center(4); pos(32)] ----
    for (int c = lane; c < NCOL; c += 32) {
        const int p = c >> 5, k = c & (KP - 1);
        const int n = nbase + p;
        const float4 ctr = *(const float4*)(x + ((size_t)b * NN + n) * 4);
        if (k < KNB) {
            const int m = knn[((size_t)b * NN + n) * KNB + k];
            const float4 nb = *(const float4*)(x + ((size_t)b * NN + m) * 4);
            const float r0 = nb.x - ctr.x, r1 = nb.y - ctr.y;
            const float r2 = nb.z - ctr.z, r3 = nb.w - ctr.w;
            sFeat[0][c] = r0; sFeat[1][c] = r1; sFeat[2][c] = r2; sFeat[3][c] = r3;
            sFeat[4][c] = ctr.x; sFeat[5][c] = ctr.y; sFeat[6][c] = ctr.z; sFeat[7][c] = ctr.w;
#pragma unroll
            for (int v = 0; v < DIMV; ++v)
                sFeat[8 + v][c] = Wh[v * 4 + 0] * r0 + Wh[v * 4 + 1] * r1 +
                                  Wh[v * 4 + 2] * r2 + Wh[v * 4 + 3] * r3;
        } else {
#pragma unroll
            for (int r = 0; r < CC; ++r) sFeat[r][c] = 0.0f;
        }
    }
    __syncthreads();

    auto loadA = [&](const float* W, int ld, int rt, int ks) -> v2f {
        const float* p = W + (size_t)(rt * 16 + mrow) * ld + ks * 4 + khalf;
        v2f a; a.x = p[0]; a.y = p[1]; return a;
    };
    auto loadB = [&](const float* S, int ct, int ks) -> v2f {
        const int col = ct * 16 + mrow;
        v2f f; f.x = S[(ks * 4 + khalf) * NCOL + col];
               f.y = S[(ks * 4 + khalf + 1) * NCOL + col]; return f;
    };

    // ---- stage 2: h1 = relu(g1 * (W1 @ feature) + b1) -> sH ----
    for (int rt = 0; rt < CH / 16; ++rt)
        for (int ct = 0; ct < NCOL / 16; ++ct) {
            v8f acc = {0, 0, 0, 0, 0, 0, 0, 0};
#pragma unroll
            for (int ks = 0; ks < CC / 4; ++ks)
                acc = wmma4(loadA(W1, CC, rt, ks), loadB(&sFeat[0][0], ct, ks), acc);
            const int col = ct * 16 + mrow;
#pragma unroll
            for (int j = 0; j < 8; ++j) {
                const int o = rt * 16 + j + (lane >> 4) * 8;
                const float h = g1[o] * acc[j] + b1[o];
                sH[o][col] = h > 0.f ? h : 0.f;
            }
        }

    // ---- stage 3: v = bn(Wv @ feature), kk_raw = Wk @ feature (shared B frags) ----
    for (int rt = 0; rt < DIMV / 16; ++rt)
        for (int ct = 0; ct < NCOL / 16; ++ct) {
            v8f av = {0, 0, 0, 0, 0, 0, 0, 0};
            v8f ak = {0, 0, 0, 0, 0, 0, 0, 0};
#pragma unroll
            for (int ks = 0; ks < CC / 4; ++ks) {
                const v2f bf = loadB(&sFeat[0][0], ct, ks);
                av = wmma4(loadA(Wv, CC, rt, ks), bf, av);
                ak = wmma4(loadA(Wk, CC, rt, ks), bf, ak);
            }
            const int col = ct * 16 + mrow;
#pragma unroll
            for (int j = 0; j < 8; ++j) {
                const int o = rt * 16 + j + (lane >> 4) * 8;
                sV[o][col] = gv[o] * av[j] + bv[o];
                sK[o][col] = ak[j];
            }
        }
    __syncthreads();

    // ---- stage 4: softmax over k (20 valid cols), zero the padding ----
    {
        float* sKf = &sK[0][0];                     // lane == d channel
        for (int p = 0; p < TP; ++p) {
            const int base = lane * NCOL + p * KP;
            float mx = -3.4e38f;
#pragma unroll
            for (int k = 0; k < KNB; ++k) mx = fmaxf(mx, sKf[base + k]);
            float e[KNB]; float s = 0.f;
#pragma unroll
            for (int k = 0; k < KNB; ++k) { e[k] = __expf(sKf[base + k] - mx); s += e[k]; }
            const float inv = 1.f / s;
#pragma unroll
            for (int k = 0; k < KNB; ++k) sKf[base + k] = e[k] * inv;
#pragma unroll
            for (int k = KNB; k < KP; ++k) sKf[base + k] = 0.f;
        }
    }
    __syncthreads();

    // ---- stage 5: h2 = relu(bn(W2 @ h1)); fused masked max-pool over k -> sFQ ----
    for (int rt = 0; rt < CH / 16; ++rt)
        for (int ct = 0; ct < NCOL / 16; ++ct) {
            v8f acc = {0, 0, 0, 0, 0, 0, 0, 0};
#pragma unroll
            for (int ks = 0; ks < CH / 4; ++ks)
                acc = wmma4(loadA(W2, CH, rt, ks), loadB(&sH[0][0], ct, ks), acc);
            const int p = ct >> 1;                      // 2 col-tiles per point
            const int kcol = (ct & 1) * 16 + mrow;      // neighbor index of this column
#pragma unroll
            for (int j = 0; j < 8; ++j) {
                const int o = rt * 16 + j + (lane >> 4) * 8;
                float h = g2[o] * acc[j] + b2[o];
                h = h > 0.f ? h : 0.f;
                h = (kcol < KNB) ? h : 0.f;             // mask padded columns
                h = fmaxf(h, __shfl_xor(h, 1, 32));
                h = fmaxf(h, __shfl_xor(h, 2, 32));
                h = fmaxf(h, __shfl_xor(h, 4, 32));
                h = fmaxf(h, __shfl_xor(h, 8, 32));     // max within each 16-lane half
                if (mrow == 0) sFQ[o][p] = fmaxf(sFQ[o][p], h);
            }
        }
    __syncthreads();

    // ---- stage 6: q = relu(bn(Wq @ feature_q)) -> sQ ----
    for (int rt = 0; rt < (HEADS * DIMK) / 16; ++rt) {
        v8f acc = {0, 0, 0, 0, 0, 0, 0, 0};
#pragma unroll
        for (int ks = 0; ks < CH / 4; ++ks) {
            v2f bf; bf.x = sFQ[ks * 4 + khalf][mrow];
                    bf.y = sFQ[ks * 4 + khalf + 1][mrow];
            acc = wmma4(loadA(Wq, CH, rt, ks), bf, acc);
        }
#pragma unroll
        for (int j = 0; j < 8; ++j) {
            const int o = rt * 16 + j + (lane >> 4) * 8;
            if (mrow < TP) {
                const float qv = gq[o] * acc[j] + bq[o];
                sQ[o][mrow] = qv > 0.f ? qv : 0.f;
            }
        }
    }
    __syncthreads();

    // ---- stage 7: kv[d,v] = sum_k kk[d,k] * v[v,k]  (per point, 32x32x32) ----
    for (int p = 0; p < TP; ++p)
        for (int dt = 0; dt < 2; ++dt)
            for (int vt = 0; vt < 2; ++vt) {
                v8f acc = {0, 0, 0, 0, 0, 0, 0, 0};
#pragma unroll
                for (int ks = 0; ks < KP / 4; ++ks) {
                    const int kd = ks * 4 + khalf;
                    v2f a; a.x = sK[dt * 16 + mrow][p * KP + kd];
                           a.y = sK[dt * 16 + mrow][p * KP + kd + 1];
                    v2f f; f.x = sV[vt * 16 + mrow][p * KP + kd];
                           f.y = sV[vt * 16 + mrow][p * KP + kd + 1];
                    acc = wmma4(a, f, acc);
                }
                const int vc = vt * 16 + mrow;
#pragma unroll
                for (int j = 0; j < 8; ++j) {
                    const int d = dt * 16 + j + (lane >> 4) * 8;
                    sKV[p][d * DIMV + vc] = acc[j];
                }
            }
    __syncthreads();

    // ---- stage 8: out[h,v] = sum_d q[h,d] * kv[d,v] -> global (B,256,N) ----
    float* op = out + (size_t)b * (HEADS * DIMV) * NN;
    for (int p = 0; p < TP; ++p) {
        const int n = nbase + p;
        for (int vt = 0; vt < 2; ++vt) {
            v8f acc = {0, 0, 0, 0, 0, 0, 0, 0};
#pragma unroll
            for (int ks = 0; ks < DIMK / 4; ++ks) {
                const int kd = ks * 4 + khalf;
                v2f a;
                a.x = (mrow < HEADS) ? sQ[mrow * DIMK + kd][p] : 0.f;
                a.y = (mrow < HEADS) ? sQ[mrow * DIMK + kd + 1][p] : 0.f;
                v2f f;
                f.x = sKV[p][kd * DIMV + vt * 16 + mrow];
                f.y = sKV[p][(kd + 1) * DIMV + vt * 16 + mrow];
                acc = wmma4(a, f, acc);
            }
            if (lane < 16) {                            // rows h = j (0..7 valid)
                const int vc = vt * 16 + lane;
#pragma unroll
                for (int j = 0; j < 8; ++j)
                    op[(size_t)(j * DIMV + vc) * NN + n] = acc[j];
            }
        }
    }
}

// ---------------------------------------------------------------------------
extern "C" void kernel_launch(void* const* d_in, const int* in_sizes, int n_in,
                              void* d_out, int out_size, void* d_ws, size_t ws_size,
                              hipStream_t stream) {
    const float* x  = (const float*)d_in[0];
    const float* Wh = (const float*)d_in[1];
    const float* W1 = (const float*)d_in[2];
    const float* g1 = (const float*)d_in[3];
    const float* b1 = (const float*)d_in[4];
    const float* W2 = (const float*)d_in[5];
    const float* g2 = (const float*)d_in[6];
    const float* b2 = (const float*)d_in[7];
    const float* Wv = (const float*)d_in[8];
    const float* gv = (const float*)d_in[9];
    const float* bv = (const float*)d_in[10];
    const float* Wk = (const float*)d_in[11];
    const float* Wq = (const float*)d_in[12];
    const float* gq = (const float*)d_in[13];
    const float* bq = (const float*)d_in[14];
    float* outf = (float*)d_out;
    int*   knn  = (int*)d_ws;               // B*N*20 ints = 5.24 MB scratch

    // output #1: xyz passthrough (B,N,4)
    copy_xyz_kernel<<<(BB * NN + 255) / 256, 256, 0, stream>>>(
        (const float4*)x, (float4*)outf, BB * NN);

    // KNN indices
    knn_kernel<<<BB * (NN / 256), 256, 0, stream>>>(x, knn);

    // fused pipeline -> output #2: (B,256,N) after the xyz section
    fused_kernel<<<BB * (NN / TP), 32, 0, stream>>>(
        x, knn, Wh, W1, g1, b1, W2, g2, b2, Wv, gv, bv, Wk, Wq, gq, bq,
        outf + (size_t)BB * NN * 4);
}